// FastHadamardTransform_36240934044418
// MI455X (gfx1250) — compile-verified
//
#include <hip/hip_runtime.h>
#include <stdint.h>

// FHT 4096 = H16 (x) H16 (x) H16, exact fp32 via V_WMMA_F32_16X16X4_F32.
// One wave32 per row; row staged in a private 16KB LDS slice via async-to-LDS.

typedef float v2f __attribute__((ext_vector_type(2)));
typedef float v4f __attribute__((ext_vector_type(4)));
typedef float v8f __attribute__((ext_vector_type(8)));

#define N_COLS 4096
#define WAVES_PER_BLOCK 2
#define THREADS (WAVES_PER_BLOCK * 32)

__device__ __forceinline__ float hsgn(int a, int b) {
  // H16[a][b] = (-1)^popcount(a & b)
  return (__popc(a & b) & 1) ? -1.0f : 1.0f;
}

__global__ __launch_bounds__(THREADS) void FastHadamardTransform_wmma_kernel(
    const float* __restrict__ X, float* __restrict__ Y, int rows) {
  __shared__ float smem[WAVES_PER_BLOCK * N_COLS];

  const int lane = threadIdx.x & 31;
  const int wave = threadIdx.x >> 5;
  const int n    = lane & 15;  // B/D column index (N)
  const int half = lane >> 4;  // selects K/M sub-range per WMMA layout
  const int row  = blockIdx.x * WAVES_PER_BLOCK + wave;
  if (row >= rows) return;     // wave-uniform; EXEC stays all-ones for WMMA

  const int wbase = wave * N_COLS;                    // float index of wave's LDS slice
  const unsigned ldsByteBase = (unsigned)wbase * 4u;  // byte offset within LDS alloc
  const int kb = 2 * half;

  // ---------- async copy-in: 4096 floats = 32 x (32 lanes x b128) ----------
  {
    const float* src = X + (size_t)row * N_COLS;
#pragma unroll
    for (int i = 0; i < 32; ++i) {
      const int e = (i * 32 + lane) * 4;  // float offset, coalesced 512B per inst
      unsigned lb = ldsByteBase + (unsigned)e * 4u;
      unsigned long long ga = (unsigned long long)(uintptr_t)(src + e);
      asm volatile("global_load_async_to_lds_b128 %0, %1, off"
                   :: "v"(lb), "v"(ga) : "memory");
    }
    asm volatile("s_wait_asynccnt 0" ::: "memory");
  }

  // ---------- A operand: H16 in registers (constant across all tiles/stages) ----------
  // A 16x4 f32 layout: lane m = lane&15; VGPR0 holds K = kb, VGPR1 holds K = kb+1.
  v2f A[4];
#pragma unroll
  for (int s = 0; s < 4; ++s) {
    const int k = 4 * s + kb;
    A[s].x = hsgn(n, k);
    A[s].y = hsgn(n, k + 1);
  }

  // ---------- Stage 1: transform along j0 (addr = (j2*16+j1)*16 + j0) ----------
  for (int t = 0; t < 16; ++t) {
    const int base = wbase + (t * 16 + n) * 16;
    v2f b0 = *(const v2f*)&smem[base + 0  + kb];
    v2f b1 = *(const v2f*)&smem[base + 4  + kb];
    v2f b2 = *(const v2f*)&smem[base + 8  + kb];
    v2f b3 = *(const v2f*)&smem[base + 12 + kb];
    v8f acc = {};
    acc = __builtin_amdgcn_wmma_f32_16x16x4_f32(false, A[0], false, b0, (short)0, acc, false, false);
    acc = __builtin_amdgcn_wmma_f32_16x16x4_f32(false, A[1], false, b1, (short)0, acc, false, false);
    acc = __builtin_amdgcn_wmma_f32_16x16x4_f32(false, A[2], false, b2, (short)0, acc, false, false);
    acc = __builtin_amdgcn_wmma_f32_16x16x4_f32(false, A[3], false, b3, (short)0, acc, false, false);
    const int ob = base + 8 * half;  // D[r+8*half][n] -> contiguous 8 floats
    *(v4f*)&smem[ob]     = __builtin_shufflevector(acc, acc, 0, 1, 2, 3);
    *(v4f*)&smem[ob + 4] = __builtin_shufflevector(acc, acc, 4, 5, 6, 7);
  }

  // ---------- Stage 2: transform along j1 (addr = j2*256 + j1*16 + k0; tile t = k0) ----------
  for (int t = 0; t < 16; ++t) {
    const int base = wbase + n * 256 + t;  // u = j2 = n (column), stride 16 over j1
    v2f b0, b1, b2, b3;
    b0.x = smem[base + (0  + kb) * 16]; b0.y = smem[base + (1  + kb) * 16];
    b1.x = smem[base + (4  + kb) * 16]; b1.y = smem[base + (5  + kb) * 16];
    b2.x = smem[base + (8  + kb) * 16]; b2.y = smem[base + (9  + kb) * 16];
    b3.x = smem[base + (12 + kb) * 16]; b3.y = smem[base + (13 + kb) * 16];
    v8f acc = {};
    acc = __builtin_amdgcn_wmma_f32_16x16x4_f32(false, A[0], false, b0, (short)0, acc, false, false);
    acc = __builtin_amdgcn_wmma_f32_16x16x4_f32(false, A[1], false, b1, (short)0, acc, false, false);
    acc = __builtin_amdgcn_wmma_f32_16x16x4_f32(false, A[2], false, b2, (short)0, acc, false, false);
    acc = __builtin_amdgcn_wmma_f32_16x16x4_f32(false, A[3], false, b3, (short)0, acc, false, false);
#pragma unroll
    for (int r = 0; r < 8; ++r)
      smem[base + (r + 8 * half) * 16] = acc[r];
  }

  // ---------- Stage 3: transform along j2 (addr = j2*256 + k1*16 + k0; tile t = k1), fold 1/64 ----------
  for (int t = 0; t < 16; ++t) {
    const int base = wbase + t * 16 + n;  // u = k0 = n, stride 256 over j2
    v2f b0, b1, b2, b3;
    b0.x = smem[base + (0  + kb) * 256]; b0.y = smem[base + (1  + kb) * 256];
    b1.x = smem[base + (4  + kb) * 256]; b1.y = smem[base + (5  + kb) * 256];
    b2.x = smem[base + (8  + kb) * 256]; b2.y = smem[base + (9  + kb) * 256];
    b3.x = smem[base + (12 + kb) * 256]; b3.y = smem[base + (13 + kb) * 256];
    v8f acc = {};
    acc = __builtin_amdgcn_wmma_f32_16x16x4_f32(false, A[0], false, b0, (short)0, acc, false, false);
    acc = __builtin_amdgcn_wmma_f32_16x16x4_f32(false, A[1], false, b1, (short)0, acc, false, false);
    acc = __builtin_amdgcn_wmma_f32_16x16x4_f32(false, A[2], false, b2, (short)0, acc, false, false);
    acc = __builtin_amdgcn_wmma_f32_16x16x4_f32(false, A[3], false, b3, (short)0, acc, false, false);
#pragma unroll
    for (int r = 0; r < 8; ++r)
      smem[base + (r + 8 * half) * 256] = acc[r] * 0.015625f;  // 1 / 2^(12/2)
  }

  // ---------- async copy-out: LDS -> global (DS and ASYNC queues are unordered: fence DS first) ----------
  asm volatile("s_wait_dscnt 0" ::: "memory");
  {
    float* dst = Y + (size_t)row * N_COLS;
#pragma unroll
    for (int i = 0; i < 32; ++i) {
      const int e = (i * 32 + lane) * 4;
      unsigned lb = ldsByteBase + (unsigned)e * 4u;
      unsigned long long ga = (unsigned long long)(uintptr_t)(dst + e);
      asm volatile("global_store_async_from_lds_b128 %0, %1, off"
                   :: "v"(ga), "v"(lb) : "memory");
    }
    asm volatile("s_wait_asynccnt 0" ::: "memory");
  }
}

extern "C" void kernel_launch(void* const* d_in, const int* in_sizes, int n_in,
                              void* d_out, int out_size, void* d_ws, size_t ws_size,
                              hipStream_t stream) {
  const float* X = (const float*)d_in[0];
  float* Y = (float*)d_out;
  const int rows = in_sizes[0] / N_COLS;  // 8192
  dim3 grid((rows + WAVES_PER_BLOCK - 1) / WAVES_PER_BLOCK);
  dim3 block(THREADS);
  FastHadamardTransform_wmma_kernel<<<grid, block, 0, stream>>>(X, Y, rows);
}